// Flow_68015102099543
// MI455X (gfx1250) — compile-verified
//
#include <hip/hip_runtime.h>

// CDNA5 / gfx1250, wave32. Conv3x3(pad=1) as GEMM: M=O=16, K=144, N=B*H*W.
// One wave computes a 6-row x 16-pixel strip of output (6 WMMA tiles) with
// 3 rotating input row-slices in registers. fp32 V_WMMA_F32_16X16X4_F32,
// K = flow*16 + c. Border strips use a branchless clamp+select gather path.

typedef __attribute__((ext_vector_type(2))) float v2f;
typedef __attribute__((ext_vector_type(8))) float v8f;

#define HH 450
#define WW 480
#define HW (HH * WW)            // 216000
#define TW 16
#define TILES_X (WW / TW)       // 30
#define RSTRIP 6
#define STRIPS_Y (HH / RSTRIP)  // 75
#define BAT 4
#define NSTRIPS (BAT * STRIPS_Y * TILES_X)  // 9000
#define WPB 8                   // 8 wave32 per 256-thread block

__global__ __launch_bounds__(256) void conv3x3_wmma_f32(
    const float* __restrict__ x,      // [B][16][H][W]
    const float* __restrict__ comb,   // [16][144], K = flow*16 + c
    const float* __restrict__ bias,   // [16]
    float* __restrict__ out)          // [B][16][H][W]
{
    const int lane = threadIdx.x & 31;
    const int wave = threadIdx.x >> 5;
    const int strip = blockIdx.x * WPB + wave;

    const int tx  = strip % TILES_X;
    const int sy  = (strip / TILES_X) % STRIPS_Y;
    const int tb  = strip / (TILES_X * STRIPS_Y);
    const int th0 = sy * RSTRIP;
    const int w0  = tx * TW;

    const int n = lane & 15;   // B column (pixel) / A row (out-channel)
    const int t = lane >> 4;   // selects K pair {2t,2t+1} of each K-quad

    const float* xb = x + tb * (16 * HW);
    float* ob = out + tb * (16 * HW) + th0 * WW + (w0 + n) + (8 * t) * HW;

    // Bias for the 8 out-channels this lane stores (o = r2 + 8t).
    float bv[8];
#pragma unroll
    for (int r2 = 0; r2 < 8; ++r2) bv[r2] = bias[r2 + 8 * t];

    // Preload all A fragments once per strip: lane row m = lane&15,
    // A[j] = comb[m][4j+2t .. 4j+2t+1]  (contiguous 8B, immediate offsets).
    v2f A[36];
    {
        const float* pa = comb + n * 144 + 2 * t;
#pragma unroll
        for (int j = 0; j < 36; ++j) A[j] = *(const v2f*)(pa + 4 * j);
    }

    // Wave-uniform interior test: strip needs rows th0-1..th0+6 and
    // columns w0-1..w0+16 in-bounds.
    const bool interior =
        (tx >= 1) && (tx <= TILES_X - 2) && (sy >= 1) && (sy <= STRIPS_Y - 2);

    if (interior) {
        // Per-lane base pointers, one per channel-quad k: &x[b, 4k+2t, th0, w0+n].
        // All B loads are then base + compile-time immediate.
        const float* baseB[4];
#pragma unroll
        for (int k = 0; k < 4; ++k)
            baseB[k] = xb + (4 * k + 2 * t) * HW + th0 * WW + (w0 + n);

        // Rotating row-slices: Bf[slot][dx][k], slot = (y - th0 + 1) % 3.
        v2f Bf[3][3][4];

        auto loadSlice = [&](int s, int yoff) {
#pragma unroll
            for (int k = 0; k < 4; ++k) {
#pragma unroll
                for (int dx = 0; dx < 3; ++dx) {
                    const float* p = baseB[k] + (yoff * WW + dx - 1);
                    v2f v;
                    v.x = p[0];      // channel 4k+2t
                    v.y = p[HW];     // channel 4k+2t+1
                    Bf[s][dx][k] = v;
                }
            }
        };

        loadSlice(0, -1);
        loadSlice(1, 0);
        loadSlice(2, 1);

#pragma unroll
        for (int r = 0; r < RSTRIP; ++r) {
            v8f acc0 = {}, acc1 = {};
#pragma unroll
            for (int j = 0; j < 36; ++j) {
                const int flow = j >> 2;
                const int dyi  = flow / 3;          // 0..2  (dy+1)
                const int dxi  = flow % 3;          // 0..2  (dx+1)
                const int k    = j & 3;
                const int slot = (r + dyi) % 3;     // slice holding y = th0+r+dyi-1
                if (j & 1)
                    acc1 = __builtin_amdgcn_wmma_f32_16x16x4_f32(
                        false, A[j], false, Bf[slot][dxi][k], (short)0, acc1, false, false);
                else
                    acc0 = __builtin_amdgcn_wmma_f32_16x16x4_f32(
                        false, A[j], false, Bf[slot][dxi][k], (short)0, acc0, false, false);
            }
#pragma unroll
            for (int r2 = 0; r2 < 8; ++r2)
                ob[r2 * HW + r * WW] = acc0[r2] + acc1[r2] + bv[r2];

            // Slot r%3 held y = th0+r-1 (dead now); refill with y = th0+r+2.
            if (r < RSTRIP - 1) loadSlice(r % 3, r + 2);
        }
    } else {
        // Border strips (~9% of waves): BRANCHLESS gather path. Clamp the
        // coordinate into the image, load unconditionally from the (valid)
        // clamped address, then select 0 for out-of-image taps. No EXEC
        // manipulation -> loads can be hoisted and clause-batched.
#pragma unroll
        for (int r = 0; r < RSTRIP; ++r) {
            const int y0 = th0 + r;

            // Per-flow (dy,dx) clamped per-lane base pointers for channel 2t,
            // plus validity masks. j-loads add (j&3)*4*HW on top.
            const float* basef[9];
            bool okf[9];
#pragma unroll
            for (int f = 0; f < 9; ++f) {
                const int dy = f / 3 - 1;
                const int dx = f % 3 - 1;
                const int y  = y0 + dy;
                const int xw = w0 + n + dx;
                okf[f] = ((unsigned)y < (unsigned)HH) &&
                         ((unsigned)xw < (unsigned)WW);
                const int yc  = y  < 0 ? 0 : (y  > HH - 1 ? HH - 1 : y);
                const int xwc = xw < 0 ? 0 : (xw > WW - 1 ? WW - 1 : xw);
                basef[f] = xb + (2 * t) * HW + yc * WW + xwc;
            }

            v8f acc0 = {}, acc1 = {};
#pragma unroll
            for (int j = 0; j < 36; ++j) {
                const int f = j >> 2;
                const int k = j & 3;
                const float* p = basef[f] + k * 4 * HW;  // channel 4k+2t
                const float vx = p[0];
                const float vy = p[HW];
                v2f bf;
                bf.x = okf[f] ? vx : 0.0f;
                bf.y = okf[f] ? vy : 0.0f;
                if (j & 1)
                    acc1 = __builtin_amdgcn_wmma_f32_16x16x4_f32(
                        false, A[j], false, bf, (short)0, acc1, false, false);
                else
                    acc0 = __builtin_amdgcn_wmma_f32_16x16x4_f32(
                        false, A[j], false, bf, (short)0, acc0, false, false);
            }
#pragma unroll
            for (int r2 = 0; r2 < 8; ++r2)
                ob[r2 * HW + r * WW] = acc0[r2] + acc1[r2] + bv[r2];
        }
    }
}

extern "C" void kernel_launch(void* const* d_in, const int* in_sizes, int n_in,
                              void* d_out, int out_size, void* d_ws, size_t ws_size,
                              hipStream_t stream) {
    const float* x    = (const float*)d_in[0];
    const float* comb = (const float*)d_in[1];
    const float* bias = (const float*)d_in[2];
    float* outp       = (float*)d_out;

    // 9000 strips (4 batches x 75 row-strips x 30 col-tiles), 8 waves/block.
    dim3 grid(NSTRIPS / WPB);  // 1125
    conv3x3_wmma_f32<<<grid, 256, 0, stream>>>(x, comb, bias, outp);
}